// GCNEncoder_30760555774417
// MI455X (gfx1250) — compile-verified
//
#include <hip/hip_runtime.h>
#include <hip/hip_bf16.h>

typedef __attribute__((ext_vector_type(2))) float    v2f;
typedef __attribute__((ext_vector_type(8))) float    v8f;
typedef __attribute__((ext_vector_type(4))) unsigned v4u;
typedef __attribute__((ext_vector_type(8))) unsigned v8u;

// ---------------- utility / init kernels ----------------

__global__ void k_fill(float* __restrict__ p, float v, int n) {
    int i = blockIdx.x * blockDim.x + threadIdx.x;
    if (i < n) p[i] = v;
}

// out[i*64 + f] = b2[f]
__global__ void k_init_out(float* __restrict__ out, const float* __restrict__ b2, int n) {
    int i = blockIdx.x * blockDim.x + threadIdx.x;
    if (i < n) out[i] = b2[i & 63];
}

// deg[dst[e]] += w[e]   (deg pre-initialized to 1.0 to account for self-loops)
__global__ void k_deg_accum(const int* __restrict__ dst, const float* __restrict__ w,
                            float* __restrict__ deg, int E) {
    int e = blockIdx.x * blockDim.x + threadIdx.x;
    if (e < E) atomicAdd(&deg[dst[e]], w[e]);
}

// deg[i] -> rsqrt(max(deg,1e-12)) in place (becomes dinv)
__global__ void k_dinv(float* __restrict__ deg, int n) {
    int i = blockIdx.x * blockDim.x + threadIdx.x;
    if (i < n) {
        float d = deg[i];
        deg[i] = (d > 0.0f) ? rsqrtf(fmaxf(d, 1e-12f)) : 0.0f;
    }
}

// ---------------- WMMA GEMM 1: h = x @ W1 (TDM-staged B) ----------------
// x: [N,256] row-major, W1: [256,128] row-major, h: [N,128].
// Grid must satisfy gridDim.x * (blockDim.x/32) == N/16 exactly (no divergent
// exits: __syncthreads inside, and WMMA requires EXEC all-ones).
// W1 is staged into LDS in two 128x128 (64KB) chunks by the Tensor Data Mover.
__global__ void k_gemm1_wmma(const float* __restrict__ x, const float* __restrict__ W,
                             float* __restrict__ h) {
    __shared__ float wlds[128 * 128];          // 64KB: one K-chunk of W1

    const int wave  = (int)((blockIdx.x * blockDim.x + threadIdx.x) >> 5);
    const int lane  = threadIdx.x & 31;
    const int m0    = wave * 16;
    const int mrow  = m0 + (lane & 15);
    const int khalf = lane >> 4;               // 0: K={0,1}, 1: K={2,3}
    const int ncol  = lane & 15;

    v8f zero = {};
    v8f acc[8];
#pragma unroll
    for (int t = 0; t < 8; ++t) acc[t] = zero;

    const float* xr = x + (size_t)mrow * 256;

    for (int c = 0; c < 2; ++c) {              // two 128-row K chunks
        __syncthreads();                       // previous chunk fully consumed
        if ((threadIdx.x >> 5) == 0) {         // wave 0 drives the TDM
            // ---- Tensor DMA descriptor (D#): 2D tile 128 cols x 128 rows, f32 ----
            unsigned long long ga =
                (unsigned long long)(size_t)(W + (size_t)c * 128 * 128);
            unsigned lds_addr = (unsigned)(size_t)(void*)wlds; // LDS byte offset
            v4u g0;
            g0[0] = 1u;                                   // count=1, user descriptor
            g0[1] = lds_addr;                             // lds_addr
            g0[2] = (unsigned)(ga & 0xFFFFFFFFu);         // global_addr[31:0]
            g0[3] = (unsigned)((ga >> 32) & 0x01FFFFFFu)  // global_addr[56:32]
                    | 0x80000000u;                        // type=2 ("image")
            v8u g1;
            g1[0] = 2u << 16;        // data_size=2 (4 bytes)
            g1[1] = 128u << 16;      // tensor_dim0[15:0]=128 in bits[63:48]
            g1[2] = 128u << 16;      // tensor_dim0 hi=0; tensor_dim1[15:0]=128
            g1[3] = 128u << 16;      // tensor_dim1 hi=0; tile_dim0=128
            g1[4] = 128u;            // tile_dim1=128; tile_dim2=0
            g1[5] = 128u;            // tensor_dim0_stride[31:0]=128
            g1[6] = 0u;              // stride hi / dim1_stride lo
            g1[7] = 0u;
            asm volatile("tensor_load_to_lds %0, %1" :: "s"(g0), "s"(g1) : "memory");
            __builtin_amdgcn_s_wait_tensorcnt(0);
        }
        __syncthreads();                       // chunk visible to all waves

        for (int k0 = 0; k0 < 128; k0 += 4) {
            const int kk = k0 + 2 * khalf;
            // A fragment from global: x[mrow][c*128 + kk .. +1] (b64 load)
            v2f a = *(const v2f*)(xr + c * 128 + kk);
#pragma unroll
            for (int t = 0; t < 8; ++t) {
                v2f b;
                b.x = wlds[kk * 128 + t * 16 + ncol];        // ds_load_b32
                b.y = wlds[(kk + 1) * 128 + t * 16 + ncol];
                acc[t] = __builtin_amdgcn_wmma_f32_16x16x4_f32(
                    false, a, false, b, (short)0, acc[t], false, false);
            }
        }
    }
    // D layout: lane l, vgpr v -> row m0 + v + 8*khalf, col t*16 + ncol
#pragma unroll
    for (int t = 0; t < 8; ++t) {
        float* orow = h + (size_t)(m0 + 8 * khalf) * 128 + t * 16 + ncol;
#pragma unroll
        for (int v = 0; v < 8; ++v)
            orow[(size_t)v * 128] = acc[t][v];
    }
}

// ---------------- WMMA GEMM 2: h2 = relu(agg + b1) @ W2 ----------------
// agg: [N,128], b1: [128], W2: [128,64] (32KB, async-staged to LDS), h2: [N,64].
// Same exact-fit grid requirement as k_gemm1_wmma.
__global__ void k_gemm2_wmma(const float* __restrict__ agg, const float* __restrict__ b1,
                             const float* __restrict__ W, float* __restrict__ h2) {
    __shared__ float wlds[128 * 64];           // 32KB: all of W2

    // ---- stage W2 with async global->LDS copies (ASYNCcnt path) ----
    {
        const unsigned ldsbase = (unsigned)(size_t)(void*)wlds;
        for (int i = threadIdx.x; i < (128 * 64) / 4; i += blockDim.x) {
            unsigned la = ldsbase + (unsigned)i * 16u;
            unsigned long long ga =
                (unsigned long long)(size_t)((const char*)W + (size_t)i * 16);
            asm volatile("global_load_async_to_lds_b128 %0, %1, off"
                         :: "v"(la), "v"(ga) : "memory");
        }
        asm volatile("s_wait_asynccnt 0x0" ::: "memory");
    }
    __syncthreads();

    const int wave  = (int)((blockIdx.x * blockDim.x + threadIdx.x) >> 5);
    const int lane  = threadIdx.x & 31;
    const int m0    = wave * 16;
    const int mrow  = m0 + (lane & 15);
    const int khalf = lane >> 4;
    const int ncol  = lane & 15;

    v8f zero = {};
    v8f acc[4];
#pragma unroll
    for (int t = 0; t < 4; ++t) acc[t] = zero;

    const float* ar = agg + (size_t)mrow * 128;

    for (int k0 = 0; k0 < 128; k0 += 4) {
        const int kk = k0 + 2 * khalf;
        v2f a;
        a.x = fmaxf(ar[kk]     + b1[kk],     0.0f);   // fused bias + ReLU
        a.y = fmaxf(ar[kk + 1] + b1[kk + 1], 0.0f);
#pragma unroll
        for (int t = 0; t < 4; ++t) {
            v2f b;
            b.x = wlds[kk * 64 + t * 16 + ncol];
            b.y = wlds[(kk + 1) * 64 + t * 16 + ncol];
            acc[t] = __builtin_amdgcn_wmma_f32_16x16x4_f32(
                false, a, false, b, (short)0, acc[t], false, false);
        }
    }
#pragma unroll
    for (int t = 0; t < 4; ++t) {
        float* orow = h2 + (size_t)(m0 + 8 * khalf) * 64 + t * 16 + ncol;
#pragma unroll
        for (int v = 0; v < 8; ++v)
            orow[(size_t)v * 64] = acc[t][v];
    }
}

// ---------------- edge scatter, layer 1 (128 features) ----------------
// One wave per edge; lane handles float4 -> 32 lanes cover the 512B feature row.
__global__ void k_scatter1(const int* __restrict__ ei, const float* __restrict__ w,
                           const float* __restrict__ dinv, const float* __restrict__ h,
                           float* __restrict__ agg, int E, int N) {
    const long wid = (long)blockIdx.x * (blockDim.x >> 5) + (threadIdx.x >> 5);
    if (wid >= (long)E + N) return;
    const int lane = threadIdx.x & 31;
    int s, d; float wt;
    if (wid < E) {                 // real edge
        s  = ei[wid];
        d  = ei[(long)E + wid];
        wt = w[wid];
    } else {                       // self loop, weight 1
        s = d = (int)(wid - E);
        wt = 1.0f;
    }
    const float norm = dinv[s] * wt * dinv[d];
    const float4 hv = *(const float4*)(h + (size_t)s * 128 + lane * 4);
    float* ap = agg + (size_t)d * 128 + lane * 4;
    atomicAdd(ap + 0, norm * hv.x);
    atomicAdd(ap + 1, norm * hv.y);
    atomicAdd(ap + 2, norm * hv.z);
    atomicAdd(ap + 3, norm * hv.w);
}

// ---------------- edge scatter, layer 2 (64 features) ----------------
__global__ void k_scatter2(const int* __restrict__ ei, const float* __restrict__ w,
                           const float* __restrict__ dinv, const float* __restrict__ h2,
                           float* __restrict__ out, int E, int N) {
    const long wid = (long)blockIdx.x * (blockDim.x >> 5) + (threadIdx.x >> 5);
    if (wid >= (long)E + N) return;
    const int lane = threadIdx.x & 31;
    int s, d; float wt;
    if (wid < E) {
        s  = ei[wid];
        d  = ei[(long)E + wid];
        wt = w[wid];
    } else {
        s = d = (int)(wid - E);
        wt = 1.0f;
    }
    const float norm = dinv[s] * wt * dinv[d];
    const float2 hv = *(const float2*)(h2 + (size_t)s * 64 + lane * 2);
    float* op = out + (size_t)d * 64 + lane * 2;
    atomicAdd(op + 0, norm * hv.x);
    atomicAdd(op + 1, norm * hv.y);
}

// ---------------- host-side launch ----------------

extern "C" void kernel_launch(void* const* d_in, const int* in_sizes, int n_in,
                              void* d_out, int out_size, void* d_ws, size_t ws_size,
                              hipStream_t stream) {
    const float* x  = (const float*)d_in[0];   // [N, 256]
    const int*   ei = (const int*)d_in[1];     // [2, E]
    const float* ew = (const float*)d_in[2];   // [E]
    const float* W1 = (const float*)d_in[3];   // [256, 128]
    const float* b1 = (const float*)d_in[4];   // [128]
    const float* W2 = (const float*)d_in[5];   // [128, 64]
    const float* b2 = (const float*)d_in[6];   // [64]
    float* out = (float*)d_out;                // [N, 64]

    const int E   = in_sizes[2];               // 3,200,000
    const int Fh  = in_sizes[4];               // 128
    const int Fin = in_sizes[3] / Fh;          // 256
    const int N   = in_sizes[0] / Fin;         // 100,000
    (void)n_in; (void)out_size; (void)ws_size;

    // workspace layout (floats): [deg | hbuf (reused as h2) | agg]
    float* ws   = (float*)d_ws;
    float* deg  = ws;                          // N floats
    float* hbuf = ws + (1 << 17);              // N*128 floats
    float* agg  = hbuf + (size_t)N * 128;      // N*128 floats

    const int T = 256;

    // 1) deg = 1.0 (self-loop weight), += w per incoming edge, -> rsqrt
    k_fill<<<(N + T - 1) / T, T, 0, stream>>>(deg, 1.0f, N);
    k_deg_accum<<<(E + T - 1) / T, T, 0, stream>>>(ei + E, ew, deg, E);
    k_dinv<<<(N + T - 1) / T, T, 0, stream>>>(deg, N);

    // 2) h = x @ W1  (WMMA f32; TDM stages W1 into LDS).
    //    Exact-fit grid: N/16 = 6250 strips = 625 blocks x 10 waves (320 thr).
    const int strips = N / 16;                 // 6250
    const int GT = 320, GW = GT / 32;          // 10 waves/block
    k_gemm1_wmma<<<strips / GW, GT, 0, stream>>>(x, W1, hbuf);

    // 3) agg = 0; scatter-add normalized messages (edges + self loops)
    k_fill<<<((size_t)N * 128 + T - 1) / T, T, 0, stream>>>(agg, 0.0f, N * 128);
    const long nw = (long)E + N;
    const int wpb = T / 32;
    k_scatter1<<<(int)((nw + wpb - 1) / wpb), T, 0, stream>>>(ei, ew, deg, hbuf, agg, E, N);

    // 4) h2 = relu(agg + b1) @ W2  (async-LDS staged W2; h2 reuses hbuf)
    k_gemm2_wmma<<<strips / GW, GT, 0, stream>>>(agg, b1, W2, hbuf);

    // 5) out = b2 broadcast; scatter-add layer-2 messages into d_out
    k_init_out<<<(N * 64 + T - 1) / T, T, 0, stream>>>(out, b2, N * 64);
    k_scatter2<<<(int)((nw + wpb - 1) / wpb), T, 0, stream>>>(ei, ew, deg, hbuf, out, E, N);
}